// GCNLayer_5394478924644
// MI455X (gfx1250) — compile-verified
//
#include <hip/hip_runtime.h>
#include <hip/hip_bf16.h>

#define N_NODES 50000
#define N_EDGES 800000
#define D_FEAT  96
#define OUT_DIM 96

#define TM_TILES   (N_NODES / 16)   // 3125
#define TN_TILES   (OUT_DIM / 16)   // 6
#define WAVES_PER_BLK 8
#define KSTEPS     (D_FEAT / 4)     // 24 WMMA K-steps

// Packed-B LDS layout: pair (W[k][col], W[k+1][col]) stored contiguously at
// pair-index (2j+half)*PAIR_STRIDE + col, where k = 4j + 2*half.
// PAIR_STRIDE=112 -> half-wave dword offset 224 % 64 == 32: lanes 0-15 and
// 16-31 hit disjoint 32-bank sets -> conflict-free ds_load_b64.
#define PAIR_STRIDE 112
#define LDS_PAIRS   (2 * KSTEPS * PAIR_STRIDE)   // 5376 pairs = 42 KB

typedef float v2f __attribute__((ext_vector_type(2)));
typedef float v8f __attribute__((ext_vector_type(8)));

// ---------------------------------------------------------------------------
// Kernel 1: zero the two degree arrays (must re-init every call: graph replay)
// ---------------------------------------------------------------------------
__global__ void gcn_zero_deg(float* __restrict__ deg_out,
                             float* __restrict__ deg_in, int n) {
    int i = blockIdx.x * blockDim.x + threadIdx.x;
    if (i < n) { deg_out[i] = 0.0f; deg_in[i] = 0.0f; }
}

// ---------------------------------------------------------------------------
// Kernel 2: accumulate degrees (one thread per edge, f32 atomics in L2)
// ---------------------------------------------------------------------------
__global__ void gcn_degrees(const int* __restrict__ src,
                            const int* __restrict__ dst,
                            float* __restrict__ deg_out,
                            float* __restrict__ deg_in, int E) {
    int e = blockIdx.x * blockDim.x + threadIdx.x;
    if (e < E) {
        atomicAdd(&deg_out[src[e]], 1.0f);
        atomicAdd(&deg_in[dst[e]], 1.0f);
    }
}

// ---------------------------------------------------------------------------
// Kernel 3: deg -> inv sqrt (in place), 0 where deg == 0
// ---------------------------------------------------------------------------
__global__ void gcn_rsqrt(float* __restrict__ deg_out,
                          float* __restrict__ deg_in, int n) {
    int i = blockIdx.x * blockDim.x + threadIdx.x;
    if (i < n) {
        float a = deg_out[i];
        deg_out[i] = (a > 0.0f) ? rsqrtf(a) : 0.0f;
        float c = deg_in[i];
        deg_in[i] = (c > 0.0f) ? rsqrtf(c) : 0.0f;
    }
}

// ---------------------------------------------------------------------------
// Kernel 4: h = x @ W via V_WMMA_F32_16X16X4_F32.
//  - Block = 256 threads = 8 waves; W pre-packed into LDS in B-fragment
//    order so every B frag is one aligned, bank-conflict-free ds_load_b64
//    (compiler pairs them into ds_load_2addr_b64).
//  - Wave w owns row tile tm = blockIdx*8 + w; loads its full 16x96 A panel
//    into registers once via a single base pointer + immediate offsets,
//    then the compiler interchanges the tn/k loops keeping 6 accumulators
//    live: each A frag feeds 6 back-to-back WMMAs.
//  A frag (16x4 f32): lanes 0-15 -> (m=lane, k={0,1}), lanes 16-31 -> k={2,3}
//  B frag (4x16 f32): same layout with N in place of M.
//  C/D (16x16 f32, 8 VGPRs): vgpr r -> row r + 8*half, col = lane&15.
// ---------------------------------------------------------------------------
__global__ void gcn_gemm_wmma(const float* __restrict__ x,
                              const float* __restrict__ W,
                              float* __restrict__ h) {
    __shared__ v2f Wl[LDS_PAIRS];

    const int tid = threadIdx.x;
    // Stage W -> LDS, packed per B-fragment: element W[k][col] with
    // k = 4j + 2*half + which goes to pair (2j+half)*112 + col, lane .x/.y.
    for (int i = tid; i < D_FEAT * OUT_DIM; i += 256) {
        const int k     = i / OUT_DIM;
        const int col   = i % OUT_DIM;
        const int j     = k >> 2;
        const int half  = (k >> 1) & 1;
        const int which = k & 1;
        float* p = reinterpret_cast<float*>(&Wl[(2 * j + half) * PAIR_STRIDE + col]);
        p[which] = W[i];
    }
    __syncthreads();

    const int wave = tid >> 5;
    const int lane = tid & 31;
    const int tm   = blockIdx.x * WAVES_PER_BLK + wave;
    if (tm >= TM_TILES) return;                 // wave-uniform: EXEC stays all-1s

    const int half = lane >> 4;                 // 0: k pair {0,1}; 1: {2,3}
    const int l    = lane & 15;

    // Single per-lane base pointer; all 24 A loads become immediate offsets.
    const float* xp = x + (size_t)(tm * 16 + l) * D_FEAT + 2 * half;
    v2f a[KSTEPS];
#pragma unroll
    for (int j = 0; j < KSTEPS; ++j) {
        a[j] = *reinterpret_cast<const v2f*>(xp + 4 * j);   // global_load_b64 offset:16*j
    }

    // Per-lane D-store base: rows/cols become immediate offsets.
    float* hp = h + (size_t)(tm * 16 + 8 * half) * OUT_DIM + l;

#pragma unroll
    for (int tn = 0; tn < TN_TILES; ++tn) {
        const int bcol = tn * 16 + l;           // B col (N) for this lane
        v8f c = {};
#pragma unroll
        for (int j = 0; j < KSTEPS; ++j) {
            const v2f b = Wl[(2 * j + half) * PAIR_STRIDE + bcol];  // ds_load_b64
            // 8 args: (neg_a, A, neg_b, B, c_mod, C, reuse_a, reuse_b)
            c = __builtin_amdgcn_wmma_f32_16x16x4_f32(
                    /*neg_a=*/false, a[j], /*neg_b=*/false, b,
                    /*c_mod=*/(short)0, c, /*reuse_a=*/false, /*reuse_b=*/false);
        }
#pragma unroll
        for (int r = 0; r < 8; ++r) {
            hp[r * OUT_DIM + tn * 16] = c[r];   // global_store_b32 imm offset
        }
    }
}

// ---------------------------------------------------------------------------
// Kernel 5: out[n, f] = b[f]  (bias broadcast doubles as zero-init of agg)
// ---------------------------------------------------------------------------
__global__ void gcn_init_out(const float* __restrict__ b,
                             float* __restrict__ out, int total) {
    int i = blockIdx.x * blockDim.x + threadIdx.x;
    if (i < total) out[i] = b[i % OUT_DIM];
}

// ---------------------------------------------------------------------------
// Kernel 6: edge scatter. 24 threads per edge, each owns a float4 slice of
// the 96-float feature row: coalesced float4 gather of h[src] (L2-resident),
// 4x global_atomic_add_f32 into out[dst] (L2-resident).
// ---------------------------------------------------------------------------
__global__ void gcn_scatter(const float* __restrict__ h,
                            const float* __restrict__ inv_out,
                            const float* __restrict__ inv_in,
                            const int* __restrict__ src,
                            const int* __restrict__ dst,
                            float* __restrict__ out, int E) {
    const int tid = blockIdx.x * blockDim.x + threadIdx.x;
    const int e   = tid / 24;
    if (e >= E) return;
    const int q = (tid % 24) * 4;

    const int s = src[e];
    const int d = dst[e];
    const float coef = inv_out[s] * inv_in[d];

    const float4 hv = *reinterpret_cast<const float4*>(h + s * OUT_DIM + q);
    float* o = out + d * OUT_DIM + q;
    atomicAdd(o + 0, hv.x * coef);
    atomicAdd(o + 1, hv.y * coef);
    atomicAdd(o + 2, hv.z * coef);
    atomicAdd(o + 3, hv.w * coef);
}

// ---------------------------------------------------------------------------
// Launch
// ---------------------------------------------------------------------------
extern "C" void kernel_launch(void* const* d_in, const int* in_sizes, int n_in,
                              void* d_out, int out_size, void* d_ws, size_t ws_size,
                              hipStream_t stream) {
    const float* x   = (const float*)d_in[0];   // [N_NODES, D_FEAT]
    const float* W   = (const float*)d_in[1];   // [D_FEAT, OUT_DIM]
    const float* b   = (const float*)d_in[2];   // [OUT_DIM]
    const int*   src = (const int*)d_in[3];     // [N_EDGES]
    const int*   dst = (const int*)d_in[4];     // [N_EDGES]
    float* out = (float*)d_out;                 // [N_NODES, OUT_DIM]

    // Workspace layout (16B-aligned offsets):
    //   h:        N_NODES * OUT_DIM floats = 19,200,000 B (offset 0)
    //   deg_out:  N_NODES floats           (offset 19,200,000)
    //   deg_in:   N_NODES floats           (offset 19,400,000)
    char* ws = (char*)d_ws;
    float* h       = (float*)(ws);
    float* deg_out = (float*)(ws + (size_t)N_NODES * OUT_DIM * sizeof(float));
    float* deg_in  = deg_out + N_NODES;

    const int BLK = 256;

    // 1) zero degree arrays
    gcn_zero_deg<<<(N_NODES + BLK - 1) / BLK, BLK, 0, stream>>>(deg_out, deg_in, N_NODES);

    // 2) degree accumulation
    gcn_degrees<<<(N_EDGES + BLK - 1) / BLK, BLK, 0, stream>>>(src, dst, deg_out, deg_in, N_EDGES);

    // 3) rsqrt normalization coefficients
    gcn_rsqrt<<<(N_NODES + BLK - 1) / BLK, BLK, 0, stream>>>(deg_out, deg_in, N_NODES);

    // 4) h = x @ W via WMMA: one wave per 16-row tile, 6 col tiles each,
    //    W packed per-fragment in LDS per block of 8 waves.
    {
        const int grid = (TM_TILES + WAVES_PER_BLK - 1) / WAVES_PER_BLK;  // 391
        gcn_gemm_wmma<<<grid, BLK, 0, stream>>>(x, W, h);
    }

    // 5) out = broadcast bias (also serves as zero-init for the scatter-add)
    {
        const int total = N_NODES * OUT_DIM;
        gcn_init_out<<<(total + BLK - 1) / BLK, BLK, 0, stream>>>(b, out, total);
    }

    // 6) edge gather/scale/scatter-add: 24 threads per edge
    {
        const long long threads = (long long)N_EDGES * 24;
        const int grid = (int)((threads + BLK - 1) / BLK);
        gcn_scatter<<<grid, BLK, 0, stream>>>(h, deg_out, deg_in, src, dst, out, N_EDGES);
    }
}